// SinkhornNormalization_75290776698915
// MI455X (gfx1250) — compile-verified
//
#include <hip/hip_runtime.h>
#include <math.h>

// Sinkhorn normalization, 8192x8192 f32, 10 iterations.
// Factored form: maintain only r[8192], c[8192] log-scaling vectors.
//   r_i = LSE_j(scores_ij - c_j);  c_j = LSE_i(scores_ij - r_i);  out = exp(scores - r - c)
// 22 streaming passes over 256MB => ~5.6GB @ 23.3TB/s ~ 0.24ms (HBM-bound).

constexpr int N              = 8192;
constexpr int TPB            = 256;
constexpr int CHUNKS         = 32;           // row chunks for the column reduction
constexpr int ROWS_PER_CHUNK = N / CHUNKS;   // 256
constexpr int NUM_ITERS      = 10;

typedef unsigned int u32x4 __attribute__((ext_vector_type(4)));
typedef int          i32x8 __attribute__((ext_vector_type(8)));
typedef int          i32x4 __attribute__((ext_vector_type(4)));

// ---------------------------------------------------------------------------
// TDM helper: DMA one contiguous row of N floats (32KB) from global into LDS.
// Builds the D# descriptor per CDNA5 ISA ch.8 (2D tensor, data_size=4B,
// tile = N x 1). Caller (one wave) must s_wait_tensorcnt + barrier.
// ---------------------------------------------------------------------------
__device__ __forceinline__ void tdm_load_row_to_lds(const float* src, void* lds_dst) {
    unsigned long long ga = (unsigned long long)(const void*)src;
    unsigned int lds_off  = (unsigned int)(size_t)lds_dst; // low 32b of generic LDS ptr = LDS offset

    // group 0: count=1 | lds_addr | global_addr[31:0] | global_addr[56:32] + type=2
    u32x4 g0 = { 1u,
                 lds_off,
                 (unsigned int)ga,
                 (((unsigned int)(ga >> 32)) & 0x01FFFFFFu) | 0x80000000u };

    // group 1: data_size=4B; tensor_dim0=tensor_dim1=N; tile = N x 1; stride0 = N.
    i32x8 g1 = { (int)(2u << 16),                                          // data_size=2 (4 bytes)
                 (int)(((unsigned)N & 0xFFFFu) << 16),                     // tensor_dim0[15:0]
                 (int)(((unsigned)N >> 16) | (((unsigned)N & 0xFFFFu) << 16)), // dim0 hi | dim1 lo
                 (int)(((unsigned)N >> 16) | ((unsigned)N << 16)),         // dim1 hi | tile_dim0=N
                 1,                                                        // tile_dim1=1, tile_dim2=0
                 N,                                                        // tensor_dim0_stride[31:0]
                 0, 0 };
    i32x4 gz = { 0, 0, 0, 0 };
#if defined(__clang_major__) && __clang_major__ >= 23
    i32x8 gz8 = { 0, 0, 0, 0, 0, 0, 0, 0 };
    __builtin_amdgcn_tensor_load_to_lds(g0, g1, gz, gz, gz8, 0);
#else
    __builtin_amdgcn_tensor_load_to_lds(g0, g1, gz, gz, 0);
#endif
}

// online logsumexp update (1 exp per element on the common path)
__device__ __forceinline__ void lse_update(float x, float& m, float& s) {
    if (x <= m) {
        s += expf(x - m);
    } else {
        s = s * expf(m - x) + 1.0f;
        m = x;
    }
}

__device__ __forceinline__ void lse_combine(float m2, float s2, float& m, float& s) {
    const float M = fmaxf(m, m2);
    s = s * expf(m - M) + s2 * expf(m2 - M);
    m = M;
}

// ---------------------------------------------------------------------------
// init: c = 0
// ---------------------------------------------------------------------------
__global__ void sk_init_c(float* __restrict__ c) {
    int j = blockIdx.x * blockDim.x + threadIdx.x;
    if (j < N) c[j] = 0.0f;
}

// ---------------------------------------------------------------------------
// Row LSE: one workgroup per row. TDM DMAs the row AND the c[] vector (32KB
// each) into LDS; 256 threads then run a b128 LDS streaming logsumexp and a
// tree reduction. No global traffic in the inner loop.
// ---------------------------------------------------------------------------
__global__ void __launch_bounds__(TPB) sk_row_lse(const float* __restrict__ scores,
                                                  const float* __restrict__ c,
                                                  float* __restrict__ r) {
    __shared__ __align__(16) float rowbuf[N];  // 32 KB
    __shared__ __align__(16) float cbuf[N];    // 32 KB
    __shared__ float red_m[TPB];
    __shared__ float red_s[TPB];

    const int tid = threadIdx.x;
    const int row = blockIdx.x;

    if (tid < 32) {  // wave 0 issues both TDM loads (EXEC-independent, TENSORcnt-tracked)
        tdm_load_row_to_lds(scores + (size_t)row * N, rowbuf);
        tdm_load_row_to_lds(c, cbuf);
        __builtin_amdgcn_s_wait_tensorcnt(0);
    }
    __syncthreads();

    // streaming logsumexp, float4 LDS reads (ds_load_b128, bank-conflict-free)
    const float4* rb4 = (const float4*)rowbuf;
    const float4* cb4 = (const float4*)cbuf;
    float m = -INFINITY, s = 0.0f;
#pragma unroll
    for (int k = 0; k < N / (TPB * 4); ++k) {   // 8 iterations
        const int j4 = tid + k * TPB;
        const float4 x = rb4[j4];
        const float4 cc = cb4[j4];
        lse_update(x.x - cc.x, m, s);
        lse_update(x.y - cc.y, m, s);
        lse_update(x.z - cc.z, m, s);
        lse_update(x.w - cc.w, m, s);
    }
    red_m[tid] = m;
    red_s[tid] = s;
    for (int off = TPB / 2; off >= 1; off >>= 1) {
        __syncthreads();
        if (tid < off) {
            float mm = red_m[tid], ss = red_s[tid];
            lse_combine(red_m[tid + off], red_s[tid + off], mm, ss);
            red_m[tid] = mm;
            red_s[tid] = ss;
        }
    }
    if (tid == 0) r[row] = red_m[0] + logf(red_s[0]);
}

// ---------------------------------------------------------------------------
// Column LSE, stage 1: each thread owns 4 consecutive columns (float4 /
// global_load_b128, 512B per wave per issue). Block covers 1024 columns x
// 256 rows; grid = (8, 32) = 256 WGs. r[] chunk pre-staged in LDS.
// ---------------------------------------------------------------------------
__global__ void __launch_bounds__(TPB) sk_col_partial(const float* __restrict__ scores,
                                                      const float* __restrict__ r,
                                                      float* __restrict__ pm,
                                                      float* __restrict__ ps) {
    __shared__ float rsh[ROWS_PER_CHUNK];
    const int tid = threadIdx.x;
    const int col = (blockIdx.x * TPB + tid) * 4;
    const int r0  = blockIdx.y * ROWS_PER_CHUNK;

    for (int i = tid; i < ROWS_PER_CHUNK; i += TPB) rsh[i] = r[r0 + i];
    __syncthreads();

    const float* p = scores + (size_t)r0 * N + col;
    float m0 = -INFINITY, m1 = -INFINITY, m2 = -INFINITY, m3 = -INFINITY;
    float s0 = 0.0f, s1 = 0.0f, s2 = 0.0f, s3 = 0.0f;
#pragma unroll 4
    for (int i = 0; i < ROWS_PER_CHUNK; ++i) {
        const float4 x = *(const float4*)(p + (size_t)i * N);
        const float ri = rsh[i];
        lse_update(x.x - ri, m0, s0);
        lse_update(x.y - ri, m1, s1);
        lse_update(x.z - ri, m2, s2);
        lse_update(x.w - ri, m3, s3);
    }
    *(float4*)(pm + (size_t)blockIdx.y * N + col) = make_float4(m0, m1, m2, m3);
    *(float4*)(ps + (size_t)blockIdx.y * N + col) = make_float4(s0, s1, s2, s3);
}

// ---------------------------------------------------------------------------
// Column LSE, stage 2: combine the 32 per-chunk partials per column.
// ---------------------------------------------------------------------------
__global__ void __launch_bounds__(TPB) sk_col_finalize(const float* __restrict__ pm,
                                                       const float* __restrict__ ps,
                                                       float* __restrict__ c) {
    const int j = blockIdx.x * blockDim.x + threadIdx.x;
    float m = pm[j], s = ps[j];
#pragma unroll
    for (int k = 1; k < CHUNKS; ++k) {
        lse_combine(pm[(size_t)k * N + j], ps[(size_t)k * N + j], m, s);
    }
    c[j] = m + logf(s);
}

// ---------------------------------------------------------------------------
// Final: out = exp(scores - r_i - c_j), float4 in / float4 out.
// ---------------------------------------------------------------------------
__global__ void __launch_bounds__(TPB) sk_exp_out(const float* __restrict__ scores,
                                                  const float* __restrict__ r,
                                                  const float* __restrict__ c,
                                                  float* __restrict__ out) {
    const int row = blockIdx.y;
    const int j   = (blockIdx.x * TPB + threadIdx.x) * 4;
    const float ri = r[row];
    const float4 sc = *(const float4*)(scores + (size_t)row * N + j);
    const float4 cc = *(const float4*)(c + j);
    float4 o;
    o.x = expf(sc.x - ri - cc.x);
    o.y = expf(sc.y - ri - cc.y);
    o.z = expf(sc.z - ri - cc.z);
    o.w = expf(sc.w - ri - cc.w);
    *(float4*)(out + (size_t)row * N + j) = o;
}

// ---------------------------------------------------------------------------
extern "C" void kernel_launch(void* const* d_in, const int* in_sizes, int n_in,
                              void* d_out, int out_size, void* d_ws, size_t ws_size,
                              hipStream_t stream) {
    (void)in_sizes; (void)n_in; (void)out_size; (void)ws_size;

    const float* scores = (const float*)d_in[0];
    float* out = (float*)d_out;

    float* ws = (float*)d_ws;
    float* r  = ws;                  // 8192
    float* c  = ws + N;              // 8192
    float* pm = ws + 2 * N;          // 32 * 8192
    float* ps = pm + CHUNKS * N;     // 32 * 8192   (total ~2.1 MB)

    sk_init_c<<<N / TPB, TPB, 0, stream>>>(c);

    for (int it = 0; it < NUM_ITERS; ++it) {
        sk_row_lse<<<N, TPB, 0, stream>>>(scores, c, r);
        sk_col_partial<<<dim3(N / (TPB * 4), CHUNKS), TPB, 0, stream>>>(scores, r, pm, ps);
        sk_col_finalize<<<N / TPB, TPB, 0, stream>>>(pm, ps, c);
    }

    sk_exp_out<<<dim3(N / (TPB * 4), N), TPB, 0, stream>>>(scores, r, c, out);
}